// Recompose_3590592660138
// MI455X (gfx1250) — compile-verified
//
#include <hip/hip_runtime.h>
#include <hip/hip_bf16.h>

typedef __attribute__((ext_vector_type(2))) float v2f;
typedef __attribute__((ext_vector_type(4))) float v4f;
typedef __attribute__((ext_vector_type(8))) float v8f;

// Kernel 1: w[k] = temporal_bases[batch_indices[1], k] * scaling_factors[k]
// (only column 1 of the reference matmul survives the [:, 1:2] slice, so the
//  whole problem is a GEMV against this one scaled basis row)
__global__ void Recompose_prep_kernel(const float* __restrict__ tb,
                                      const float* __restrict__ sc,
                                      const long long* __restrict__ idx,
                                      float* __restrict__ w, int K) {
    int k = blockIdx.x * blockDim.x + threadIdx.x;
    long long r = idx[1];
    if (k < K) {
        w[k] = tb[r * (long long)K + k] * sc[k];
    }
}

// Kernel 2: out[i] = sum_k targets[i,k] * w[k], via V_WMMA_F32_16X16X4_F32.
// One wave per 16-row tile; w staged in LDS once per block (B operand comes
// from ds_load_b64 broadcasts); two independent WMMA accumulator chains for ILP.
//
// A-matrix layout (ISA 7.12.2, 32-bit A 16x4):
//   lanes 0-15 : M = lane,    VGPR0 = K=k0,   VGPR1 = K=k0+1
//   lanes 16-31: M = lane-16, VGPR0 = K=k0+2, VGPR1 = K=k0+3
// B (4x16) = w[k0..k0+3] broadcast across all 16 columns:
//   lanes 0-15 : VGPR0 = w[k0],   VGPR1 = w[k0+1]
//   lanes 16-31: VGPR0 = w[k0+2], VGPR1 = w[k0+3]
// D (16x16 f32): lane<16 -> N=lane, VGPRv -> M=v ; lane>=16 -> M=v+8.
// All columns of D are identical, so lane 0 stores rows 0-7, lane 16 rows 8-15.
__global__ __launch_bounds__(256) void Recompose_gemv_wmma_kernel(
    const float* __restrict__ targets,
    const float* __restrict__ w,
    float* __restrict__ out,
    int K, int Brows) {
    extern __shared__ float ws[];   // K floats (4 KB for K=1024)

    const int tid  = threadIdx.x;
    const int lane = tid & 31;
    const int wave = tid >> 5;

    // Stage w into LDS: 256 threads x float4 covers K=1024 in one pass.
    for (int k = tid * 4; k < K; k += blockDim.x * 4) {
        *(v4f*)(ws + k) = *(const v4f*)(w + k);
    }
    __syncthreads();

    const int tile    = blockIdx.x * 8 + wave;  // one 16-row tile per wave
    const int rowbase = tile * 16;
    if (rowbase >= Brows) return;               // wave-uniform (launch is exact anyway)

    const int half = lane >> 4;                 // 0: K-pair {0,1}, 1: K-pair {2,3}
    const int m    = lane & 15;                 // row within tile
    const float* arow = targets + (size_t)(rowbase + m) * (size_t)K + (half << 1);
    const float* brow = ws + (half << 1);       // LDS pointer -> ds_load_b64

    v8f c0 = {0.f, 0.f, 0.f, 0.f, 0.f, 0.f, 0.f, 0.f};
    v8f c1 = {0.f, 0.f, 0.f, 0.f, 0.f, 0.f, 0.f, 0.f};

    // K assumed multiple of 32 (K = 1024). Two independent accumulation
    // chains (c0: first 16 K of each 32-span, c1: second 16) for WMMA ILP.
    for (int k0 = 0; k0 < K; k0 += 32) {
        v2f a00 = *(const v2f*)(arow + k0);
        v2f a01 = *(const v2f*)(arow + k0 + 4);
        v2f a02 = *(const v2f*)(arow + k0 + 8);
        v2f a03 = *(const v2f*)(arow + k0 + 12);
        v2f a10 = *(const v2f*)(arow + k0 + 16);
        v2f a11 = *(const v2f*)(arow + k0 + 20);
        v2f a12 = *(const v2f*)(arow + k0 + 24);
        v2f a13 = *(const v2f*)(arow + k0 + 28);

        v2f b00 = *(const v2f*)(brow + k0);
        v2f b01 = *(const v2f*)(brow + k0 + 4);
        v2f b02 = *(const v2f*)(brow + k0 + 8);
        v2f b03 = *(const v2f*)(brow + k0 + 12);
        v2f b10 = *(const v2f*)(brow + k0 + 16);
        v2f b11 = *(const v2f*)(brow + k0 + 20);
        v2f b12 = *(const v2f*)(brow + k0 + 24);
        v2f b13 = *(const v2f*)(brow + k0 + 28);

        // 8 args: (neg_a, A, neg_b, B, c_mod, C, reuse_a, reuse_b)
        c0 = __builtin_amdgcn_wmma_f32_16x16x4_f32(false, a00, false, b00, (short)0, c0, false, false);
        c1 = __builtin_amdgcn_wmma_f32_16x16x4_f32(false, a10, false, b10, (short)0, c1, false, false);
        c0 = __builtin_amdgcn_wmma_f32_16x16x4_f32(false, a01, false, b01, (short)0, c0, false, false);
        c1 = __builtin_amdgcn_wmma_f32_16x16x4_f32(false, a11, false, b11, (short)0, c1, false, false);
        c0 = __builtin_amdgcn_wmma_f32_16x16x4_f32(false, a02, false, b02, (short)0, c0, false, false);
        c1 = __builtin_amdgcn_wmma_f32_16x16x4_f32(false, a12, false, b12, (short)0, c1, false, false);
        c0 = __builtin_amdgcn_wmma_f32_16x16x4_f32(false, a03, false, b03, (short)0, c0, false, false);
        c1 = __builtin_amdgcn_wmma_f32_16x16x4_f32(false, a13, false, b13, (short)0, c1, false, false);
    }

    v8f c = c0 + c1;

    // Columns of D are identical; lane 0 holds M=0..7, lane 16 holds M=8..15.
    if (m == 0) {
        float* o = out + rowbase + half * 8;
#pragma unroll
        for (int v = 0; v < 8; ++v) o[v] = c[v];
    }
}

extern "C" void kernel_launch(void* const* d_in, const int* in_sizes, int n_in,
                              void* d_out, int out_size, void* d_ws, size_t ws_size,
                              hipStream_t stream) {
    const float*     targets = (const float*)d_in[0];      // (B, K) f32
    const float*     tb      = (const float*)d_in[1];      // (T, K) f32
    const float*     sc      = (const float*)d_in[2];      // (K,)  f32
    const long long* idx     = (const long long*)d_in[3];  // (B,)  int64
    float*           out     = (float*)d_out;              // (B, 1) f32
    float*           w       = (float*)d_ws;               // K floats scratch

    const int K     = in_sizes[2];           // 1024
    const int Brows = in_sizes[0] / K;       // 8192

    Recompose_prep_kernel<<<(K + 255) / 256, 256, 0, stream>>>(tb, sc, idx, w, K);

    const int tiles  = (Brows + 15) / 16;        // 512
    const int blocks = (tiles + 7) / 8;          // 64 blocks x 8 waves
    const size_t lds = (size_t)K * sizeof(float);
    Recompose_gemv_wmma_kernel<<<blocks, 256, lds, stream>>>(targets, w, out, K, Brows);
}